// REINFORCE_9019431321799
// MI455X (gfx1250) — compile-verified
//
#include <hip/hip_runtime.h>
#include <hip/hip_bf16.h>
#include <stdint.h>

// ---------------------------------------------------------------------------
// REINFORCE objective + discounted reverse cumsum for MI455X (gfx1250).
// B = T = 4096, fp32. Memory-bound (~320MB @ 23.3 TB/s ~ 14us floor).
// Scan via 16x16 triangular-matrix WMMA (V_WMMA_F32_16X16X4_F32 x4) +
// log-depth decayed suffix scan in LDS. Input staging via
// GLOBAL_LOAD_ASYNC_TO_LDS_B128 (ASYNCcnt / s_wait_asynccnt).
// Gamma powers via hardware exp2 (v_exp_f32, via __builtin_amdgcn_exp2f)
// instead of runtime-indexed tables (which lowered to huge v_cndmask
// cascades in round 1).
// ---------------------------------------------------------------------------

typedef float v2f __attribute__((ext_vector_type(2)));
typedef float v8f __attribute__((ext_vector_type(8)));

#define GAMMA_F    0.99f
#define CLIP_F     5.0f
#define BB         4096
#define TT         4096
#define SEG        16
#define NSEG       (TT / SEG)       // 256 segments per row
#define THREADS    256
#define ROWCH      32               // row chunks for column reduction
#define ROWS_PER_CH (BB / ROWCH)    // 128

// log2(0.99), double-rounded to float
#define LOG2G      (-0.014499569695115089f)

__host__ __device__ constexpr float gpow(int n) {
  float r = 1.0f;
  for (int i = 0; i < n; ++i) r *= GAMMA_F;
  return r;
}

__device__ __forceinline__ float gamma_pow(int d) {
  // gamma^d via hardware 2^x (v_exp_f32); ~1 ulp, no table, no select cascade
  return __builtin_amdgcn_exp2f((float)d * LOG2G);
}

__device__ __forceinline__ uint32_t lds_off(const void* p) {
  // low 32 bits of the flat address of a __shared__ object == LDS byte offset
  return (uint32_t)(uintptr_t)p;
}

__device__ __forceinline__ void async_ld_b128(uint32_t lds_dst, const void* gsrc) {
  // CDNA5 async global->LDS copy (tracked by ASYNCcnt, no VGPR data path)
  asm volatile("global_load_async_to_lds_b128 %0, %1, off"
               :: "v"(lds_dst), "v"(gsrc) : "memory");
}

__device__ __forceinline__ void wait_async0() {
  asm volatile("s_wait_asynccnt 0x0" ::: "memory");
}

// ---------------------------------------------------------------------------
// Pass 1: per-row discounted reverse cumsum.
//   wr = weight * log_sigmoid(logits)
//   cum[t] = wr[t] + g * cum[t+1]
// One 256-thread block (8 waves) per row. Each wave owns 512 elements
// = 2 chunks of 16 segments x 16 elements. Local scans via WMMA:
//   C = U x WR,  U[i][k] = g^(k-i) (k>=i), done as 4 chained 16x16x4 WMMAs.
// Cross-segment carries: Hillis-Steele decayed suffix scan over 256 entries.
// ---------------------------------------------------------------------------
__global__ __launch_bounds__(THREADS)
void scan_kernel(const float* __restrict__ logits,
                 const float* __restrict__ weight,
                 float* __restrict__ cum) {
  __shared__ __align__(16) float wrbuf[TT];   // logits staging -> wr values
  __shared__ __align__(16) float cbuf[TT];    // weight staging -> local scans
  __shared__ float Sbuf[2][NSEG];             // segment-scan ping-pong

  const int t   = threadIdx.x;
  const int row = blockIdx.x;
  const float* lrow = logits + (size_t)row * TT;
  const float* wrow = weight + (size_t)row * TT;

  // --- stage logits & weight into LDS via async tensor path (coalesced) ---
  #pragma unroll
  for (int v = 0; v < 4; ++v) {
    const int e = v * 1024 + t * 4;
    async_ld_b128(lds_off(&wrbuf[e]), lrow + e);
    async_ld_b128(lds_off(&cbuf[e]),  wrow + e);
  }
  wait_async0();
  __syncthreads();

  // --- wr = weight * log_sigmoid(logits), in place over wrbuf ---
  #pragma unroll
  for (int v = 0; v < 4; ++v) {
    const int e = v * 1024 + t * 4;
    #pragma unroll
    for (int q = 0; q < 4; ++q) {
      const float x  = wrbuf[e + q];
      const float ls = fminf(x, 0.0f) - log1pf(__expf(-fabsf(x)));  // stable
      wrbuf[e + q] = cbuf[e + q] * ls;
    }
  }
  __syncthreads();

  // --- WMMA local scans: C = U x WR per 256-element chunk ---
  const int lane = t & 31;
  const int wv   = t >> 5;        // wave index 0..7
  const int j    = lane & 15;     // column (= A row m, = B col n)
  const int h    = lane >> 4;     // lane half

  // A operand (16x4 slices of U). f32 A layout: lanes 0-15 M=0..15,
  // VGPR0 holds K = {0 | 2} per half, VGPR1 holds K = {1 | 3}.
  // U[j][k] = gamma^(k-j) for k >= j, else 0. Built with 8 hardware exp2s.
  v2f Am[4];
  #pragma unroll
  for (int p = 0; p < 4; ++p) {
    const int k0 = 4 * p + 2 * h;
    Am[p].x = (k0     >= j) ? gamma_pow(k0     - j) : 0.0f;
    Am[p].y = (k0 + 1 >= j) ? gamma_pow(k0 + 1 - j) : 0.0f;
  }

  #pragma unroll
  for (int c = 0; c < 2; ++c) {
    const int base = wv * 512 + c * 256;
    // B operand (4x16 slices of WR): lane holds column j, K split {2h, 2h+1}
    v2f Bv[4];
    #pragma unroll
    for (int p = 0; p < 4; ++p) {
      const float2 d = *(const float2*)&wrbuf[base + 16 * j + 4 * p + 2 * h];
      Bv[p].x = d.x;
      Bv[p].y = d.y;
    }
    v8f acc = {};
    #pragma unroll
    for (int p = 0; p < 4; ++p) {
      acc = __builtin_amdgcn_wmma_f32_16x16x4_f32(
          /*neg_a=*/false, Am[p], /*neg_b=*/false, Bv[p],
          /*c_mod=*/(short)0, acc, /*reuse_a=*/false, /*reuse_b=*/false);
    }
    // D layout: VGPR r -> M = r + 8h (position in segment), N = j (segment)
    #pragma unroll
    for (int r = 0; r < 8; ++r)
      cbuf[base + 16 * j + (r + 8 * h)] = acc[r];
  }
  __syncthreads();

  // --- decayed suffix scan over 256 segment-start values, factor g^16 ---
  Sbuf[0][t] = cbuf[16 * t];
  __syncthreads();
  float f  = gpow(SEG);   // gamma^16 (constexpr); squares each step
  int  cur = 0;
  #pragma unroll
  for (int s = 1; s < NSEG; s <<= 1) {
    const float v = Sbuf[cur][t] + ((t + s < NSEG) ? f * Sbuf[cur][t + s] : 0.0f);
    Sbuf[cur ^ 1][t] = v;
    __syncthreads();
    cur ^= 1;
    f *= f;
  }
  const float* Sf = Sbuf[cur];  // Sf[s] = C_s (full scan at segment start)

  // --- apply carries, coalesced float4 store of cum ---
  // i0 = 4*(t&3) is loop-invariant; carry factors gamma^(16-i0-q) computed
  // once: one exp2 + three multiplies by 1/gamma.
  const int   i0   = 4 * (t & 3);
  const float ginv = 1.0f / GAMMA_F;
  const float f0   = gamma_pow(16 - i0);
  const float f1   = f0 * ginv;
  const float f2   = f1 * ginv;
  const float f3   = f2 * ginv;

  float* outRow = cum + (size_t)row * TT;
  #pragma unroll
  for (int v = 0; v < 4; ++v) {
    const int e = v * 1024 + t * 4;
    const int s = e >> 4;          // segment index (same for all 4 elems)
    const float carry = (s + 1 < NSEG) ? Sf[s + 1] : 0.0f;
    float4 o;
    o.x = cbuf[e + 0] + f0 * carry;
    o.y = cbuf[e + 1] + f1 * carry;
    o.z = cbuf[e + 2] + f2 * carry;
    o.w = cbuf[e + 3] + f3 * carry;
    *(float4*)&outRow[e] = o;
  }
}

// ---------------------------------------------------------------------------
// Pass 2a: fixed-order partial column sums of (cum - baselines).
// Grid (T/256, ROWCH). Deterministic (no atomics).
// ---------------------------------------------------------------------------
__global__ __launch_bounds__(THREADS)
void colsum_partial(const float* __restrict__ cum,
                    const float* __restrict__ baselines,
                    float* __restrict__ partial) {
  const int col = blockIdx.x * THREADS + threadIdx.x;
  const int r0  = blockIdx.y * ROWS_PER_CH;
  float s = 0.0f;
  for (int r = 0; r < ROWS_PER_CH; ++r) {
    const size_t idx = (size_t)(r0 + r) * TT + col;
    s += cum[idx] - baselines[idx];
  }
  partial[(size_t)blockIdx.y * TT + col] = s;
}

// Pass 2b: reduce ROWCH partials -> column mean.
__global__ __launch_bounds__(THREADS)
void colmean_final(const float* __restrict__ partial,
                   float* __restrict__ mean) {
  const int col = blockIdx.x * THREADS + threadIdx.x;
  float s = 0.0f;
  #pragma unroll
  for (int y = 0; y < ROWCH; ++y) s += partial[(size_t)y * TT + col];
  mean[col] = s * (1.0f / (float)BB);
}

// ---------------------------------------------------------------------------
// Pass 3: objective[b] = sum_t clip(cum - baseline - mean) * log_probs.
// One block per row; wave32 shuffle reduction + LDS cross-wave combine.
// ---------------------------------------------------------------------------
__global__ __launch_bounds__(THREADS)
void objective_kernel(const float* __restrict__ cum,
                      const float* __restrict__ baselines,
                      const float* __restrict__ logp,
                      const float* __restrict__ mean,
                      float* __restrict__ obj) {
  __shared__ float red[THREADS / 32];
  const int t   = threadIdx.x;
  const int row = blockIdx.x;
  const size_t roff = (size_t)row * TT;

  float acc = 0.0f;
  #pragma unroll
  for (int v = 0; v < 4; ++v) {
    const int e = v * 1024 + t * 4;
    const float4 c4 = *(const float4*)&cum[roff + e];
    const float4 b4 = *(const float4*)&baselines[roff + e];
    const float4 p4 = *(const float4*)&logp[roff + e];
    const float4 m4 = *(const float4*)&mean[e];
    float a;
    a = fminf(fmaxf(c4.x - b4.x - m4.x, -CLIP_F), CLIP_F); acc += a * p4.x;
    a = fminf(fmaxf(c4.y - b4.y - m4.y, -CLIP_F), CLIP_F); acc += a * p4.y;
    a = fminf(fmaxf(c4.z - b4.z - m4.z, -CLIP_F), CLIP_F); acc += a * p4.z;
    a = fminf(fmaxf(c4.w - b4.w - m4.w, -CLIP_F), CLIP_F); acc += a * p4.w;
  }
  // wave32 reduction (explicit width 32 — CDNA5 is wave32-only)
  #pragma unroll
  for (int off = 16; off >= 1; off >>= 1) acc += __shfl_down(acc, off, 32);
  if ((t & 31) == 0) red[t >> 5] = acc;
  __syncthreads();
  if (t == 0) {
    float s = 0.0f;
    #pragma unroll
    for (int w = 0; w < THREADS / 32; ++w) s += red[w];
    obj[row] = s;
  }
}

// ---------------------------------------------------------------------------
extern "C" void kernel_launch(void* const* d_in, const int* in_sizes, int n_in,
                              void* d_out, int out_size, void* d_ws, size_t ws_size,
                              hipStream_t stream) {
  const float* log_probs = (const float*)d_in[0];
  const float* logits    = (const float*)d_in[1];
  const float* weight    = (const float*)d_in[2];
  const float* baselines = (const float*)d_in[3];

  float* obj = (float*)d_out;            // [B]
  float* cum = (float*)d_out + BB;       // [B,T] (second tuple output)

  float* partial = (float*)d_ws;                       // [ROWCH, T]
  float* mean    = (float*)d_ws + (size_t)ROWCH * TT;  // [T]
  // ws usage: (ROWCH+1)*T*4 = 540,672 bytes

  scan_kernel<<<BB, THREADS, 0, stream>>>(logits, weight, cum);
  colsum_partial<<<dim3(TT / THREADS, ROWCH), THREADS, 0, stream>>>(cum, baselines, partial);
  colmean_final<<<TT / THREADS, THREADS, 0, stream>>>(partial, mean);
  objective_kernel<<<BB, THREADS, 0, stream>>>(cum, baselines, log_probs, mean, obj);
}